// Propagate_Planes_Layer_8856222564796
// MI455X (gfx1250) — compile-verified
//
#include <hip/hip_runtime.h>
#include <math.h>
#include <stdint.h>

typedef __attribute__((ext_vector_type(2))) float v2f;
typedef __attribute__((ext_vector_type(8))) float v8f;

#define FFT_N   2048
#define PI_F    3.14159265358979323846f
#define TWO_PI_F 6.28318530717958647692f
#define GRID_DX 1.0e-6f
#define DIST_Z  1.0e-3f
#define NIMG    6              // 3 wavelengths x 2 batch
#define WAVES_PER_BLOCK 4

#define DEVINL __attribute__((always_inline)) static __device__ inline

// ---------------------------------------------------------------------------
// f32 WMMA: D(16x16) = A(16x4) * B(4x16) + C   (exact fp32 on the matrix pipe)
// ---------------------------------------------------------------------------
DEVINL v8f wmma_f32_k4(v2f a, v2f b, v8f c) {
  return __builtin_amdgcn_wmma_f32_16x16x4_f32(false, a, false, b, (short)0, c,
                                               false, false);
}

// ---------------------------------------------------------------------------
// Async global->LDS staging of one contiguous 2048-float row (8 KB).
// Per-lane B128 = 16 bytes; 512 B/instr/wave; 16 instructions per row.
// ---------------------------------------------------------------------------
DEVINL void async_row_to_lds(const float* g, float* lds, int lane) {
  uint32_t lbase = (uint32_t)(uintptr_t)lds;   // low 32 bits == LDS offset
  uint64_t gbase = (uint64_t)(uintptr_t)g;
  uint32_t off = (uint32_t)lane * 16u;
#pragma unroll
  for (int it = 0; it < 16; ++it) {
    asm volatile("global_load_async_to_lds_b128 %0, %1, %2"
                 :: "v"(lbase + off), "v"(off), "s"(gbase)
                 : "memory");
    off += 512u;
  }
}
DEVINL void wait_async0() {
  asm volatile("s_wait_asynccnt 0" ::: "memory");
}

// Fast sin/cos for arguments already reduced to [0, 2*pi): no slow path.
DEVINL void fast_sincos(float th, float* s, float* c) {
  *s = __sinf(th);
  *c = __cosf(th);
}

// ---------------------------------------------------------------------------
// 2048-point FFT executed by ONE wave32, data in wave-private LDS.
// Mixed radix 16*16*8:
//   stage A: DFT16 across stride-128 groups   (WMMA, 8 tiles, 16 wmma each)
//   stage B: DFT16 across stride-8 groups     (WMMA, 8 tiles, 16 wmma each)
//   stage C: DFT8 + digit-reversed scatter    (VALU)
// DIR=+1: forward (exp(-i)), DIR=-1: inverse (exp(+i), scaled by 1/N).
// A-layout (16x4 f32): lanes0-15 M=lane, vgpr0/1=K0/K1; lanes16-31 K2/K3.
// B-layout (4x16 f32): lanes0-15 N=lane, vgpr0/1=K0/K1; lanes16-31 K2/K3.
// D-layout: vgpr j -> M=j (lanes<16) / M=j+8 (lanes>=16), N=lane%16.
// ---------------------------------------------------------------------------
template <int DIR>
DEVINL void fft2048_wave(float* Ar, float* Ai, float* Br, float* Bi,
                         int lane) {
  const int m  = lane & 15;
  const int hi = lane >> 4;

  // DFT-16 matrix W[m][k] = exp(-i*2pi*m*k/16) -> C=cos, S=sin components.
  // (m*k) mod 16 keeps the angle in [0, 2*pi): fast trig path only.
  v2f C[4], S[4];
  {
    const float step = TWO_PI_F / 16.0f;
#pragma unroll
    for (int kc = 0; kc < 4; ++kc) {
      int k0 = 4 * kc + 2 * hi;
      float a0 = step * (float)((m * k0) & 15);
      float a1 = step * (float)((m * (k0 + 1)) & 15);
      C[kc].x = __cosf(a0); C[kc].y = __cosf(a1);
      S[kc].x = __sinf(a0); S[kc].y = __sinf(a1);
    }
  }

  // ---------------- stage A: radix-16 over a, n = a*128 + r ----------------
  for (int t = 0; t < 8; ++t) {
    const int col = 16 * t + m;
    v2f br[4], bi[4];
#pragma unroll
    for (int kc = 0; kc < 4; ++kc) {
      int kk = 4 * kc + 2 * hi;
      br[kc].x = Ar[kk * 128 + col];       br[kc].y = Ar[(kk + 1) * 128 + col];
      bi[kc].x = Ai[kk * 128 + col];       bi[kc].y = Ai[(kk + 1) * 128 + col];
    }
    v8f dr = {}, di = {};
#pragma unroll
    for (int kc = 0; kc < 4; ++kc) {
      v2f p = (DIR > 0) ? S[kc] : -S[kc];  // multiplies Xi into Yr
      v2f q = -p;                          // multiplies Xr into Yi
      dr = wmma_f32_k4(C[kc], br[kc], dr);
      dr = wmma_f32_k4(p,     bi[kc], dr);
      di = wmma_f32_k4(q,     br[kc], di);
      di = wmma_f32_k4(C[kc], bi[kc], di);
    }
#pragma unroll
    for (int j = 0; j < 8; ++j) {          // twiddle w2048^(k1*r), store
      int k1 = j + 8 * hi;
      float th = (TWO_PI_F / (float)FFT_N) * (float)((k1 * col) & (FFT_N - 1));
      float c, s;
      fast_sincos(th, &s, &c);
      s = (DIR > 0) ? -s : s;
      float yr = dr[j], yi = di[j];
      Ar[k1 * 128 + col] = yr * c - yi * s;
      Ai[k1 * 128 + col] = yr * s + yi * c;
    }
  }

  // ------------- stage B: radix-16 over b, r = b*8 + s (per k1-row) --------
  for (int t = 0; t < 8; ++t) {            // tile batches rows 2t and 2t+1
    const int row = 2 * t + (m >> 3);
    const int s8i = m & 7;
    v2f br[4], bi[4];
#pragma unroll
    for (int kc = 0; kc < 4; ++kc) {
      int kk = 4 * kc + 2 * hi;
      br[kc].x = Ar[row * 128 + kk * 8 + s8i];
      br[kc].y = Ar[row * 128 + (kk + 1) * 8 + s8i];
      bi[kc].x = Ai[row * 128 + kk * 8 + s8i];
      bi[kc].y = Ai[row * 128 + (kk + 1) * 8 + s8i];
    }
    v8f dr = {}, di = {};
#pragma unroll
    for (int kc = 0; kc < 4; ++kc) {
      v2f p = (DIR > 0) ? S[kc] : -S[kc];
      v2f q = -p;
      dr = wmma_f32_k4(C[kc], br[kc], dr);
      dr = wmma_f32_k4(p,     bi[kc], dr);
      di = wmma_f32_k4(q,     br[kc], di);
      di = wmma_f32_k4(C[kc], bi[kc], di);
    }
#pragma unroll
    for (int j = 0; j < 8; ++j) {          // twiddle w128^(k2*s), store
      int k2 = j + 8 * hi;
      float th = (TWO_PI_F / 128.0f) * (float)((k2 * s8i) & 127);
      float c, s;
      fast_sincos(th, &s, &c);
      s = (DIR > 0) ? -s : s;
      float yr = dr[j], yi = di[j];
      Ar[row * 128 + k2 * 8 + s8i] = yr * c - yi * s;
      Ai[row * 128 + k2 * 8 + s8i] = yr * s + yi * c;
    }
  }

  // ------------- stage C: radix-8 over s + digit-reversed scatter ----------
  const float scale = (DIR > 0) ? 1.0f : (1.0f / (float)FFT_N);
  constexpr float W8C[8] = {1.0f, 0.70710678f, 0.0f, -0.70710678f,
                            -1.0f, -0.70710678f, 0.0f, 0.70710678f};
  constexpr float W8S[8] = {0.0f, 0.70710678f, 1.0f, 0.70710678f,
                            0.0f, -0.70710678f, -1.0f, -0.70710678f};
  for (int t = 0; t < 8; ++t) {
    const int idx = lane + 32 * t;         // 256 independent DFT-8s
    const int k1 = idx & 15;
    const int k2 = idx >> 4;
    const int base = k1 * 128 + k2 * 8;
    float xr[8], xi[8];
#pragma unroll
    for (int s = 0; s < 8; ++s) { xr[s] = Ar[base + s]; xi[s] = Ai[base + s]; }
#pragma unroll
    for (int k3 = 0; k3 < 8; ++k3) {
      float yr = 0.0f, yi = 0.0f;
#pragma unroll
      for (int s = 0; s < 8; ++s) {
        const int j = (s * k3) & 7;
        const float wr = W8C[j];
        const float wi = (DIR > 0) ? -W8S[j] : W8S[j];
        yr += xr[s] * wr - xi[s] * wi;
        yi += xr[s] * wi + xi[s] * wr;
      }
      const int n = k1 + 16 * k2 + 256 * k3;   // natural-order output
      Br[n] = yr * scale;
      Bi[n] = yi * scale;
    }
  }
}

// ---------------------------------------------------------------------------
// Pass 1: field build + forward row FFT -> planar workspace
// ---------------------------------------------------------------------------
__global__ __launch_bounds__(128) void pass1_rows(
    const float* __restrict__ amp, const float* __restrict__ ph,
    const float* __restrict__ ap, float* __restrict__ wsRe,
    float* __restrict__ wsIm) {
  extern __shared__ float sm[];
  const int wid  = threadIdx.x >> 5;
  const int lane = threadIdx.x & 31;
  float* Ar = sm + wid * 4 * FFT_N;
  float* Ai = Ar + FFT_N;
  float* Br = Ai + FFT_N;
  float* Bi = Br + FFT_N;
  const int row = blockIdx.x * WAVES_PER_BLOCK + wid;   // [0, NIMG*N)
  const int y = row & (FFT_N - 1);
  const size_t rbase = (size_t)row * FFT_N;

  async_row_to_lds(amp + rbase, Br, lane);   // stage amp into Br
  async_row_to_lds(ph + rbase, Bi, lane);    // stage phase into Bi
  wait_async0();

  for (int i = lane; i < FFT_N; i += 32) {
    float a = Br[i] * ap[y * FFT_N + i];
    float sp, cp;
    sincosf(Bi[i], &sp, &cp);                // unbounded arg: full-range trig
    Ar[i] = a * cp;
    Ai[i] = a * sp;
  }
  fft2048_wave<1>(Ar, Ai, Br, Bi, lane);
  for (int i = lane; i < FFT_N; i += 32) {
    wsRe[rbase + i] = Br[i];
    wsIm[rbase + i] = Bi[i];
  }
}

// ---------------------------------------------------------------------------
// Pass 2: forward col FFT, xH(fx,fy,lambda), inverse col FFT (fused in LDS)
// ---------------------------------------------------------------------------
__global__ __launch_bounds__(128) void pass2_cols(float* __restrict__ wsRe,
                                                  float* __restrict__ wsIm) {
  extern __shared__ float sm[];
  const int wid  = threadIdx.x >> 5;
  const int lane = threadIdx.x & 31;
  float* Ar = sm + wid * 4 * FFT_N;
  float* Ai = Ar + FFT_N;
  float* Br = Ai + FFT_N;
  float* Bi = Br + FFT_N;
  const int cidx = blockIdx.x * WAVES_PER_BLOCK + wid;  // [0, NIMG*N)
  const int img = cidx >> 11;
  const int x = cidx & (FFT_N - 1);
  const int w = img >> 1;                               // wavelength index
  const float lam = (w == 0) ? 4.0e-7f : ((w == 1) ? 5.32e-7f : 7.0e-7f);
  const size_t ibase = (size_t)img * FFT_N * FFT_N;

  for (int yy = lane; yy < FFT_N; yy += 32) {
    Ar[yy] = wsRe[ibase + (size_t)yy * FFT_N + x];
    Ai[yy] = wsIm[ibase + (size_t)yy * FFT_N + x];
  }
  fft2048_wave<1>(Ar, Ai, Br, Bi, lane);

  const float finv = 1.0f / ((float)FFT_N * GRID_DX);
  const float fx = (float)((x < FFT_N / 2) ? x : x - FFT_N) * finv;
  for (int k = lane; k < FFT_N; k += 32) {
    float fy = (float)((k < FFT_N / 2) ? k : k - FFT_N) * finv;
    float arg = 1.0f - lam * lam * (fx * fx + fy * fy);
    float hr = 0.0f, hx = 0.0f;
    if (arg > 0.0f) {
      float kz = (TWO_PI_F / lam) * sqrtf(arg);
      sincosf(kz * DIST_Z, &hx, &hr);        // large arg: full-range trig
    }
    float br = Br[k], bi = Bi[k];
    Br[k] = br * hr - bi * hx;
    Bi[k] = br * hx + bi * hr;
  }
  fft2048_wave<-1>(Br, Bi, Ar, Ai, lane);
  for (int yy = lane; yy < FFT_N; yy += 32) {
    wsRe[ibase + (size_t)yy * FFT_N + x] = Ar[yy];
    wsIm[ibase + (size_t)yy * FFT_N + x] = Ai[yy];
  }
}

// ---------------------------------------------------------------------------
// Pass 3: inverse row FFT + abs/angle -> outputs
// ---------------------------------------------------------------------------
__global__ __launch_bounds__(128) void pass3_rows(
    const float* __restrict__ wsRe, const float* __restrict__ wsIm,
    float* __restrict__ outAbs, float* __restrict__ outAng) {
  extern __shared__ float sm[];
  const int wid  = threadIdx.x >> 5;
  const int lane = threadIdx.x & 31;
  float* Ar = sm + wid * 4 * FFT_N;
  float* Ai = Ar + FFT_N;
  float* Br = Ai + FFT_N;
  float* Bi = Br + FFT_N;
  const int row = blockIdx.x * WAVES_PER_BLOCK + wid;
  const size_t rbase = (size_t)row * FFT_N;

  async_row_to_lds(wsRe + rbase, Ar, lane);
  async_row_to_lds(wsIm + rbase, Ai, lane);
  wait_async0();

  fft2048_wave<-1>(Ar, Ai, Br, Bi, lane);
  for (int i = lane; i < FFT_N; i += 32) {
    float re = Br[i], im = Bi[i];
    outAbs[rbase + i] = sqrtf(re * re + im * im);
    outAng[rbase + i] = atan2f(im, re);
  }
}

// ---------------------------------------------------------------------------
extern "C" void kernel_launch(void* const* d_in, const int* in_sizes, int n_in,
                              void* d_out, int out_size, void* d_ws,
                              size_t ws_size, hipStream_t stream) {
  (void)in_sizes; (void)n_in; (void)out_size; (void)ws_size;
  const float* amp = (const float*)d_in[0];
  const float* ph  = (const float*)d_in[1];
  const float* ap  = (const float*)d_in[2];
  float* wsRe = (float*)d_ws;
  float* wsIm = wsRe + (size_t)NIMG * FFT_N * FFT_N;
  float* outAbs = (float*)d_out;
  float* outAng = outAbs + (size_t)NIMG * FFT_N * FFT_N;

  const int blocks = (NIMG * FFT_N) / WAVES_PER_BLOCK;           // 3072
  const int tpb = 32 * WAVES_PER_BLOCK;                          // 128
  const size_t lds = (size_t)WAVES_PER_BLOCK * 4 * FFT_N * sizeof(float); // 128KB

  hipFuncSetAttribute((const void*)pass1_rows,
                      hipFuncAttributeMaxDynamicSharedMemorySize, (int)lds);
  hipFuncSetAttribute((const void*)pass2_cols,
                      hipFuncAttributeMaxDynamicSharedMemorySize, (int)lds);
  hipFuncSetAttribute((const void*)pass3_rows,
                      hipFuncAttributeMaxDynamicSharedMemorySize, (int)lds);

  pass1_rows<<<blocks, tpb, lds, stream>>>(amp, ph, ap, wsRe, wsIm);
  pass2_cols<<<blocks, tpb, lds, stream>>>(wsRe, wsIm);
  pass3_rows<<<blocks, tpb, lds, stream>>>(wsRe, wsIm, outAbs, outAng);
}